// SimpleNN_4355096838715
// MI455X (gfx1250) — compile-verified
//
#include <hip/hip_runtime.h>
#include <hip/hip_bf16.h>
#include <cmath>

typedef __attribute__((ext_vector_type(2))) float v2f;
typedef __attribute__((ext_vector_type(8))) float v8f;

#define TPB 256

// ---------------- helpers ----------------

__device__ __forceinline__ float atomicMaxF(float* addr, float v) {
    // int ordering matches float ordering for v>=0; uint ordering is reversed for v<0
    if (v >= 0.0f) {
        return __int_as_float(atomicMax((int*)addr, __float_as_int(v)));
    } else {
        return __uint_as_float(atomicMin((unsigned int*)addr, (unsigned int)__float_as_int(v)));
    }
}

__global__ void k_fill(float* __restrict__ p, float v, size_t n) {
    size_t i = (size_t)blockIdx.x * blockDim.x + threadIdx.x;
    if (i < n) p[i] = v;
}

// ---------------- layer 1 projections (K=3, scalar) ----------------
__global__ void k_proj1(const float* __restrict__ x,
                        const float* __restrict__ Wq, const float* __restrict__ bq,
                        const float* __restrict__ Wk, const float* __restrict__ bk,
                        const float* __restrict__ Wv, const float* __restrict__ bv,
                        const float* __restrict__ Ws, const float* __restrict__ bs,
                        float* __restrict__ q, float* __restrict__ k,
                        float* __restrict__ v, float* __restrict__ skip, int N) {
    int i = blockIdx.x * blockDim.x + threadIdx.x;
    if (i >= N * 64) return;
    int n = i >> 6, c = i & 63;
    float x0 = x[n * 3 + 0], x1 = x[n * 3 + 1], x2 = x[n * 3 + 2];
    q[i]    = x0 * Wq[c] + x1 * Wq[64 + c] + x2 * Wq[128 + c] + bq[c];
    k[i]    = x0 * Wk[c] + x1 * Wk[64 + c] + x2 * Wk[128 + c] + bk[c];
    v[i]    = x0 * Wv[c] + x1 * Wv[64 + c] + x2 * Wv[128 + c] + bv[c];
    skip[i] = x0 * Ws[c] + x1 * Ws[64 + c] + x2 * Ws[128 + c] + bs[c];
}

// ---------------- edge pass A: logits + segment max ----------------
__global__ void k_edge_logits(const int* __restrict__ ei,
                              const float* __restrict__ qb, const float* __restrict__ kb,
                              int stride, int D, float scale, int E,
                              float* __restrict__ wlog, float* __restrict__ m) {
    int i = blockIdx.x * blockDim.x + threadIdx.x;
    if (i >= 2 * E) return;
    int e = i >> 1, h = i & 1;
    int src = ei[e], dst = ei[E + e];
    const float4* qr = (const float4*)(qb + (size_t)dst * stride + h * D);
    const float4* kr = (const float4*)(kb + (size_t)src * stride + h * D);
    float acc = 0.0f;
    int n4 = D >> 2;
    for (int j = 0; j < n4; ++j) {
        float4 a = qr[j], b = kr[j];
        acc += a.x * b.x + a.y * b.y + a.z * b.z + a.w * b.w;
    }
    float lg = acc * scale;
    wlog[i] = lg;
    atomicMaxF(&m[dst * 2 + h], lg);
}

// nodes with no incoming edges: m stays -inf -> reference sets it to 0
__global__ void k_fixm(float* __restrict__ m, int n2) {
    int i = blockIdx.x * blockDim.x + threadIdx.x;
    if (i >= n2) return;
    float mv = m[i];
    if (mv == -INFINITY) m[i] = 0.0f;
}

// ---------------- edge pass B1: w = exp(logit - m[dst]); s[dst] += w ----------------
__global__ void k_edge_w_s(const int* __restrict__ ei, float* __restrict__ wlog,
                           const float* __restrict__ m, float* __restrict__ s, int E) {
    int i = blockIdx.x * blockDim.x + threadIdx.x;
    if (i >= 2 * E) return;
    int e = i >> 1, h = i & 1;
    int dst = ei[E + e];
    float w = __expf(wlog[i] - m[dst * 2 + h]);
    wlog[i] = w;
    atomicAdd(&s[dst * 2 + h], w);
}

// ---------------- edge pass B2: acc[dst,c] += w * v[src,c] ----------------
__global__ void k_edge_acc(const int* __restrict__ ei, const float* __restrict__ wlog,
                           const float* __restrict__ vb, int vstride,
                           int hdshift, int dshift, float* __restrict__ acc, int E) {
    size_t i = (size_t)blockIdx.x * blockDim.x + threadIdx.x;
    size_t total = (size_t)E << hdshift;
    if (i >= total) return;
    int HD = 1 << hdshift;
    int e = (int)(i >> hdshift);
    int c = (int)i & (HD - 1);
    int h = c >> dshift;
    int src = ei[e], dst = ei[E + e];
    float w = wlog[e * 2 + h];
    atomicAdd(acc + (((size_t)dst) << hdshift) + c, w * vb[(size_t)src * vstride + c]);
}

// ---------------- node finalize: out = relu(skip + acc/(s+eps)) ----------------
__global__ void k_node_out(const float* __restrict__ acc, const float* __restrict__ s,
                           const float* __restrict__ skipb, int skipstride,
                           float* __restrict__ outb, int outstride,
                           int N, int hdshift, int dshift) {
    size_t i = (size_t)blockIdx.x * blockDim.x + threadIdx.x;
    size_t total = (size_t)N << hdshift;
    if (i >= total) return;
    int HD = 1 << hdshift;
    int n = (int)(i >> hdshift);
    int c = (int)i & (HD - 1);
    int h = c >> dshift;
    float val = skipb[(size_t)n * skipstride + c] + acc[i] / (s[n * 2 + h] + 1e-16f);
    outb[(size_t)n * outstride + c] = fmaxf(val, 0.0f);
}

// ---------------- pack fused layer-2 weights, TRANSPOSED ----------------
// WallT[512][64]: WallT[c*64 + r] = W_c(r),  c -> [Wq2|Wk2|Wv2|Ws2]
__global__ void k_pack2(const float* __restrict__ Wq, const float* __restrict__ Wk,
                        const float* __restrict__ Wv, const float* __restrict__ Ws,
                        const float* __restrict__ bq, const float* __restrict__ bk,
                        const float* __restrict__ bv, const float* __restrict__ bs,
                        float* __restrict__ WallT, float* __restrict__ ball) {
    int i = blockIdx.x * blockDim.x + threadIdx.x;
    if (i < 512 * 64) {
        int c = i >> 6, r = i & 63;
        const float* srcw = (c < 128) ? Wq : (c < 256) ? Wk : (c < 384) ? Wv : Ws;
        WallT[i] = srcw[r * 128 + (c & 127)];
    } else if (i < 512 * 64 + 512) {
        int c = i - 512 * 64;
        const float* srcb = (c < 128) ? bq : (c < 256) ? bk : (c < 384) ? bv : bs;
        ball[c] = srcb[c & 127];
    }
}

// ---------------- layer-2 projections via fp32 WMMA ----------------
// out[N,512] = h1[N,64] @ W[64,512] + ball, weights given transposed (WallT[512][64]).
// 16-node tile per block; A tile staged once into LDS via async global->LDS DMA
// (GLOBAL_LOAD_ASYNC_TO_LDS_B128), then 8 waves x 4 col-tiles x 16 WMMA_F32_16X16X4.
#define A_STRIDE 68   // 16B-aligned row stride, spreads lanes over LDS banks

__global__ void k_proj2_wmma(const float* __restrict__ h1, const float* __restrict__ WallT,
                             const float* __restrict__ ball, float* __restrict__ out, int N) {
    __shared__ float sA[16 * A_STRIDE];
    int m0 = blockIdx.x * 16;
    int t = threadIdx.x;

    // cooperative async stage: one b128 per thread -> 16 rows x 64 floats
    {
        int r = t >> 4, q4 = t & 15;                 // row, 16-byte chunk
        int row = m0 + r; if (row >= N) row = N - 1; // clamp tail reads
        const float* gsrc = h1 + (size_t)row * 64 + q4 * 4;
        unsigned lds_addr = (unsigned)(size_t)(&sA[r * A_STRIDE + q4 * 4]);
        asm volatile("global_load_async_to_lds_b128 %0, %1, off"
                     :: "v"(lds_addr), "v"(gsrc) : "memory");
    }
    asm volatile("s_wait_asynccnt 0" ::: "memory");
    __syncthreads();

    int wave = t >> 5;                 // 0..7
    int lane = t & 31;
    int lrow = lane & 15;
    int khalf = (lane >> 4) ? 2 : 0;   // A/B layout: lanes 16-31 hold K+2/K+3

    v8f c[4];
    int cols[4];
    for (int i = 0; i < 4; ++i) {
        cols[i] = (wave + 8 * i) * 16 + lrow;
        float bv = ball[cols[i]];
        for (int r = 0; r < 8; ++r) c[i][r] = bv;   // bias broadcast down each column
    }
    const float* arow = &sA[lrow * A_STRIDE];
    for (int k = 0; k < 64; k += 4) {
        v2f a = *(const v2f*)(arow + k + khalf);                       // ds_load_b64
        for (int i = 0; i < 4; ++i) {
            v2f b = *(const v2f*)(WallT + (size_t)cols[i] * 64 + k + khalf);  // global b64
            c[i] = __builtin_amdgcn_wmma_f32_16x16x4_f32(
                false, a, false, b, (short)0, c[i], false, false);
        }
    }

    int mh = (lane >> 4) * 8;   // C/D layout: lanes 16-31 hold rows M+8
    if (m0 + 16 <= N) {
        // fast path: unguarded store clause
        for (int i = 0; i < 4; ++i) {
            float* ob = out + (size_t)(m0 + mh) * 512 + cols[i];
            #pragma unroll
            for (int r = 0; r < 8; ++r) ob[(size_t)r * 512] = c[i][r];
        }
    } else {
        for (int i = 0; i < 4; ++i) {
            for (int r = 0; r < 8; ++r) {
                int mm = m0 + r + mh;
                if (mm < N) out[(size_t)mm * 512 + cols[i]] = c[i][r];
            }
        }
    }
}

// ---------------- global max pool ----------------
__global__ void k_pool(const float* __restrict__ h2, const int* __restrict__ batch,
                       float* __restrict__ g, int N) {
    size_t i = (size_t)blockIdx.x * blockDim.x + threadIdx.x;
    if (i >= (size_t)N * 128) return;
    int n = (int)(i >> 7), c = (int)i & 127;
    atomicMaxF(&g[batch[n] * 128 + c], h2[i]);
}

// ---------------- MLP head (single block) ----------------
// d_out layout: logits[64*40] then latent[64*32]
__global__ void k_head(const float* __restrict__ g,
                       const float* __restrict__ W1, const float* __restrict__ b1,
                       const float* __restrict__ W2, const float* __restrict__ b2,
                       const float* __restrict__ W3, const float* __restrict__ b3,
                       float* __restrict__ out) {
    __shared__ float lat[64 * 32];
    __shared__ float hm[64 * 128];
    int t = threadIdx.x;
    for (int i = t; i < 64 * 32; i += TPB) {
        int r = i >> 5, c = i & 31;
        float a = b1[c];
        for (int kk = 0; kk < 128; ++kk) a += g[r * 128 + kk] * W1[kk * 32 + c];
        a = fmaxf(a, 0.0f);
        lat[i] = a;
        out[2560 + i] = a;          // latent output
    }
    __syncthreads();
    for (int i = t; i < 64 * 128; i += TPB) {
        int r = i >> 7, c = i & 127;
        float a = b2[c];
        for (int kk = 0; kk < 32; ++kk) a += lat[r * 32 + kk] * W2[kk * 128 + c];
        hm[i] = fmaxf(a, 0.0f);
    }
    __syncthreads();
    for (int i = t; i < 64 * 40; i += TPB) {
        int r = i / 40, c = i % 40;
        float a = b3[c];
        for (int kk = 0; kk < 128; ++kk) a += hm[r * 128 + kk] * W3[kk * 40 + c];
        out[i] = a;                 // logits output
    }
}

// ---------------- host ----------------

static inline unsigned gblocks(size_t total) { return (unsigned)((total + TPB - 1) / TPB); }

extern "C" void kernel_launch(void* const* d_in, const int* in_sizes, int n_in,
                              void* d_out, int out_size, void* d_ws, size_t ws_size,
                              hipStream_t stream) {
    const float* x     = (const float*)d_in[0];
    const int*   ei    = (const int*)d_in[1];
    const int*   batch = (const int*)d_in[2];
    const float* Wq1 = (const float*)d_in[3],  *bq1 = (const float*)d_in[4];
    const float* Wk1 = (const float*)d_in[5],  *bk1 = (const float*)d_in[6];
    const float* Wv1 = (const float*)d_in[7],  *bv1 = (const float*)d_in[8];
    const float* Ws1 = (const float*)d_in[9],  *bs1 = (const float*)d_in[10];
    const float* Wq2 = (const float*)d_in[11], *bq2 = (const float*)d_in[12];
    const float* Wk2 = (const float*)d_in[13], *bk2 = (const float*)d_in[14];
    const float* Wv2 = (const float*)d_in[15], *bv2 = (const float*)d_in[16];
    const float* Ws2 = (const float*)d_in[17], *bs2 = (const float*)d_in[18];
    const float* W1  = (const float*)d_in[19], *b1  = (const float*)d_in[20];
    const float* W2  = (const float*)d_in[21], *b2  = (const float*)d_in[22];
    const float* W3  = (const float*)d_in[23], *b3  = (const float*)d_in[24];
    (void)n_in; (void)out_size; (void)ws_size;

    const int N = in_sizes[0] / 3;
    const int E = in_sizes[1] / 2;

    // ---- workspace arena (floats), with lifetime-based aliasing ----
    float* ws = (float*)d_ws;
    size_t o = 0;
    float* f_h1   = ws + o; o += (size_t)N * 64;   // skip1, then h1 in-place
    float* f_q1   = ws + o; o += (size_t)N * 64;
    float* f_k1   = ws + o; o += (size_t)N * 64;
    float* f_v1   = ws + o; o += (size_t)N * 64;
    float* f_acc1 = ws + o; o += (size_t)N * 64;
    float* f_wlog = ws + o; o += (size_t)E * 2;    // logits -> exp weights (both layers)
    float* f_m    = ws + o; o += (size_t)N * 2;    // shared both layers
    float* f_s    = ws + o; o += (size_t)N * 2;
    float* f_qkv2 = ws + o; o += (size_t)N * 512;  // [q2|k2|v2|skip2] interleaved per node
    float* f_g    = ws + o; o += 64 * 128;
    float* f_Wall = ws + o; o += 512 * 64;         // transposed fused W2
    float* f_ball = ws + o; o += 512;
    float* f_acc2 = f_q1;   // overlays q1+k1 (dead after edge pass A of layer 1)
    float* f_h2   = f_v1;   // overlays v1+acc1 (dead after layer-1 finalize)

    const float SC1 = 0.17677669529663687f;  // 1/sqrt(32)
    const float SC2 = 0.125f;                // 1/sqrt(64)

    // ======== layer 1 ========
    k_fill<<<gblocks((size_t)N * 2), TPB, 0, stream>>>(f_m, -INFINITY, (size_t)N * 2);
    k_fill<<<gblocks((size_t)N * 2), TPB, 0, stream>>>(f_s, 0.0f, (size_t)N * 2);
    k_fill<<<gblocks((size_t)N * 64), TPB, 0, stream>>>(f_acc1, 0.0f, (size_t)N * 64);

    k_proj1<<<gblocks((size_t)N * 64), TPB, 0, stream>>>(
        x, Wq1, bq1, Wk1, bk1, Wv1, bv1, Ws1, bs1, f_q1, f_k1, f_v1, f_h1, N);

    k_edge_logits<<<gblocks((size_t)E * 2), TPB, 0, stream>>>(
        ei, f_q1, f_k1, 64, 32, SC1, E, f_wlog, f_m);
    k_fixm<<<gblocks((size_t)N * 2), TPB, 0, stream>>>(f_m, N * 2);
    k_edge_w_s<<<gblocks((size_t)E * 2), TPB, 0, stream>>>(ei, f_wlog, f_m, f_s, E);
    k_edge_acc<<<gblocks((size_t)E * 64), TPB, 0, stream>>>(
        ei, f_wlog, f_v1, 64, 6, 5, f_acc1, E);
    k_node_out<<<gblocks((size_t)N * 64), TPB, 0, stream>>>(
        f_acc1, f_s, f_h1, 64, f_h1, 64, N, 6, 5);

    // ======== layer 2 ========
    k_pack2<<<gblocks(512 * 64 + 512), TPB, 0, stream>>>(
        Wq2, Wk2, Wv2, Ws2, bq2, bk2, bv2, bs2, f_Wall, f_ball);
    k_fill<<<gblocks((size_t)N * 2), TPB, 0, stream>>>(f_m, -INFINITY, (size_t)N * 2);
    k_fill<<<gblocks((size_t)N * 2), TPB, 0, stream>>>(f_s, 0.0f, (size_t)N * 2);
    k_fill<<<gblocks((size_t)N * 128), TPB, 0, stream>>>(f_acc2, 0.0f, (size_t)N * 128);

    k_proj2_wmma<<<(unsigned)((N + 15) / 16), TPB, 0, stream>>>(f_h1, f_Wall, f_ball, f_qkv2, N);

    k_edge_logits<<<gblocks((size_t)E * 2), TPB, 0, stream>>>(
        ei, f_qkv2 + 0, f_qkv2 + 128, 512, 64, SC2, E, f_wlog, f_m);
    k_fixm<<<gblocks((size_t)N * 2), TPB, 0, stream>>>(f_m, N * 2);
    k_edge_w_s<<<gblocks((size_t)E * 2), TPB, 0, stream>>>(ei, f_wlog, f_m, f_s, E);
    k_edge_acc<<<gblocks((size_t)E * 128), TPB, 0, stream>>>(
        ei, f_wlog, f_qkv2 + 256, 512, 7, 6, f_acc2, E);
    k_node_out<<<gblocks((size_t)N * 128), TPB, 0, stream>>>(
        f_acc2, f_s, f_qkv2 + 384, 512, f_h2, 128, N, 7, 6);

    // ======== pool + head ========
    k_fill<<<gblocks(64 * 128), TPB, 0, stream>>>(f_g, -3.402823466e38f, 64 * 128);
    k_pool<<<gblocks((size_t)N * 128), TPB, 0, stream>>>(f_h2, batch, f_g, N);
    k_head<<<1, TPB, 0, stream>>>(f_g, W1, b1, W2, b2, W3, b3, (float*)d_out);
}